// LightGCN_59090160058391
// MI455X (gfx1250) — compile-verified
//
#include <hip/hip_runtime.h>
#include <stdint.h>
#include <stddef.h>

// LightGCN eval on MI455X (gfx1250, wave32).
//   out = 0.25*emb  (init kernel)
//   out[r*128 + k] += 0.75 * v * emb[c*128 + k]  for every COO edge (r,c,v)
//
// Memory-bound scatter: emb (51MB) and out (51MB) are L2-resident (192MB L2),
// so gathers/atomics run at L2 rates; HBM sees only the 19MB edge stream.
// CDNA5 paths used: async global->LDS edge staging (ASYNCcnt), s_wait_asynccnt,
// global_prefetch_b8, per-lane b128 gathers, hardware f32 atomics.

#define D_FEAT   128
#define TILE_E   256          // edges staged per block tile
#define BLOCK_T  256          // 8 wave32 per block
#define TILE_ITERS 2          // tiles per persistent block (enables prefetch)

typedef __attribute__((address_space(1))) int gbl_int;   // global AS
typedef __attribute__((address_space(3))) int lds_int;   // LDS AS

#if defined(__has_builtin)
# if __has_builtin(__builtin_amdgcn_global_load_async_to_lds_b32)
#  define HAVE_ASYNC_LDS 1
# endif
# if __has_builtin(__builtin_amdgcn_s_wait_asynccnt)
#  define HAVE_WAIT_ASYNC 1
# endif
#endif

__device__ __forceinline__ void async_cp_b32(const void* g, void* l) {
#if defined(HAVE_ASYNC_LDS)
  // GLOBAL_LOAD_ASYNC_TO_LDS_B32: per-lane DMA global -> LDS, ASYNCcnt-tracked
  __builtin_amdgcn_global_load_async_to_lds_b32((gbl_int*)g, (lds_int*)l,
                                                /*offset=*/0, /*cpol=*/0);
#else
  *(volatile int*)l = *(const int*)g;   // synchronous fallback
#endif
}

__device__ __forceinline__ void wait_async0() {
#if defined(HAVE_ASYNC_LDS)
# if defined(HAVE_WAIT_ASYNC)
  __builtin_amdgcn_s_wait_asynccnt(0);
# else
  asm volatile("s_wait_asynccnt 0" ::: "memory");
# endif
#endif
}

// ---------------- scatter: one wave per edge, one b128 gather per lane ----
// (defined first so the disasm snippet shows the async/atomic body)
__global__ void __launch_bounds__(BLOCK_T)
lightgcn_scatter(const float* __restrict__ emb,
                 const float* __restrict__ vals,
                 const int* __restrict__ rows,
                 const int* __restrict__ cols,
                 float* __restrict__ out,
                 int n_edges, int n_tiles) {
  __shared__ int   s_rows[TILE_E];
  __shared__ int   s_cols[TILE_E];
  __shared__ float s_vals[TILE_E];

  const int tid  = threadIdx.x;      // 0..255
  const int lane = tid & 31;         // wave32 lane
  const int wave = tid >> 5;         // 0..7

  for (int t = blockIdx.x; t < n_tiles; t += gridDim.x) {
    const int base = t * TILE_E;
    const int e    = base + tid;

    // Stage this tile's edge arrays into LDS asynchronously.
    if (e < n_edges) {
      async_cp_b32(rows + e, &s_rows[tid]);
      async_cp_b32(cols + e, &s_cols[tid]);
      async_cp_b32(vals + e, &s_vals[tid]);
    }

    // Prefetch the tile this block will process next (global_prefetch_b8).
    const int np = (t + (int)gridDim.x) * TILE_E + tid;
    if (np < n_edges) {
      __builtin_prefetch(rows + np, 0, 1);
      __builtin_prefetch(cols + np, 0, 1);
      __builtin_prefetch(vals + np, 0, 1);
    }

    wait_async0();
    __syncthreads();

    const int lim = (n_edges - base < TILE_E) ? (n_edges - base) : TILE_E;

    // Wave w handles edges w, w+8, w+16, ... of the tile. Edge metadata is a
    // wave-uniform LDS broadcast; each lane gathers 4 consecutive floats of
    // emb[c] (one global_load_b128) and issues 4 no-return f32 atomics.
    #pragma unroll 4
    for (int j = wave; j < lim; j += BLOCK_T / 32) {
      const int   r = s_rows[j];
      const int   c = s_cols[j];
      const float v = 0.75f * s_vals[j];

      const float4 x =
          reinterpret_cast<const float4*>(emb)[(size_t)c * (D_FEAT / 4) + lane];

      float* dst = out + (size_t)r * D_FEAT + lane * 4;
      unsafeAtomicAdd(dst + 0, v * x.x);
      unsafeAtomicAdd(dst + 1, v * x.y);
      unsafeAtomicAdd(dst + 2, v * x.z);
      unsafeAtomicAdd(dst + 3, v * x.w);
    }
    __syncthreads();   // LDS reuse across tile iterations
  }
}

// ---------------- init: out = 0.25 * emb (16B per lane) ----------------
__global__ void lightgcn_init(const float4* __restrict__ emb4,
                              float4* __restrict__ out4, int n4) {
  int i = blockIdx.x * blockDim.x + threadIdx.x;
  if (i < n4) {
    float4 e = emb4[i];
    float4 o;
    o.x = 0.25f * e.x; o.y = 0.25f * e.y;
    o.z = 0.25f * e.z; o.w = 0.25f * e.w;
    out4[i] = o;
  }
}

extern "C" void kernel_launch(void* const* d_in, const int* in_sizes, int n_in,
                              void* d_out, int out_size, void* d_ws, size_t ws_size,
                              hipStream_t stream) {
  (void)n_in; (void)out_size; (void)d_ws; (void)ws_size;

  const float* emb  = (const float*)d_in[0];   // [N, 128] f32
  const float* vals = (const float*)d_in[1];   // [E] f32
  const int*   rows = (const int*)d_in[2];     // [E] i32
  const int*   cols = (const int*)d_in[3];     // [E] i32
  float*       out  = (float*)d_out;           // [N, 128] f32

  const int n_elem  = in_sizes[0];             // N * 128
  const int n_edges = in_sizes[1];
  const int n4      = n_elem / 4;

  // 1) out = 0.25 * emb
  {
    int grid = (n4 + BLOCK_T - 1) / BLOCK_T;
    lightgcn_init<<<grid, BLOCK_T, 0, stream>>>((const float4*)emb,
                                                (float4*)out, n4);
  }

  // 2) atomic scatter of 0.75 * v * emb[c] into out[r]
  {
    int n_tiles = (n_edges + TILE_E - 1) / TILE_E;
    int grid = (n_tiles + TILE_ITERS - 1) / TILE_ITERS;
    if (grid < 1) grid = 1;
    lightgcn_scatter<<<grid, BLOCK_T, 0, stream>>>(emb, vals, rows, cols, out,
                                                   n_edges, n_tiles);
  }
}